// Classifier_74234214744874
// MI455X (gfx1250) — compile-verified
//
#include <hip/hip_runtime.h>
#include <math.h>

// ---------------------------------------------------------------------------
// GCN classifier for MI455X (gfx1250), wave32 + V_WMMA_F32_16X16X4_F32.
// Pipeline: BN(train stats) -> [mean-agg -> GEMM(WMMA f32) -> ReLU] x2
//           -> per-graph mean pool -> FC(64->32->1) -> sigmoid.
// Mean-finalize is a separate branch-light pass so the WMMA inner loop is
// pure (b64 A-load + clause of B-loads + 4 WMMAs), no EXEC divergence.
// ---------------------------------------------------------------------------

typedef __attribute__((ext_vector_type(2))) float v2f;
typedef __attribute__((ext_vector_type(8))) float v8f;

#define N_FEATS 86
#define KPAD    88          // 86 padded to multiple of 4 for WMMA K loop
#define GCN_H   64
#define FC_H    32
#define BN_EPS  1e-5f

// ---------------- BatchNorm statistics (sum / sumsq per feature) -----------
__global__ void bn_stats_kernel(const float* __restrict__ x,
                                float* __restrict__ sums,
                                float* __restrict__ sumsq, int n_nodes) {
  __shared__ float ls[N_FEATS];
  __shared__ float lq[N_FEATS];
  int tid = threadIdx.x;
  for (int f = tid; f < N_FEATS; f += blockDim.x) { ls[f] = 0.f; lq[f] = 0.f; }
  __syncthreads();
  long long total = (long long)n_nodes * N_FEATS;
  long long per = (total + gridDim.x - 1) / (long long)gridDim.x;
  long long beg = (long long)blockIdx.x * per;
  long long end = beg + per; if (end > total) end = total;
  for (long long i = beg + tid; i < end; i += blockDim.x) {
    float v = x[i];
    int f = (int)(i % N_FEATS);
    atomicAdd(&ls[f], v);
    atomicAdd(&lq[f], v * v);
  }
  __syncthreads();
  for (int f = tid; f < N_FEATS; f += blockDim.x) {
    atomicAdd(&sums[f], ls[f]);
    atomicAdd(&sumsq[f], lq[f]);
  }
}

// --------------- fold mean/var/gamma/beta into scale & shift ---------------
__global__ void bn_finalize_kernel(const float* __restrict__ sums,
                                   const float* __restrict__ sumsq,
                                   const float* __restrict__ gamma,
                                   const float* __restrict__ beta,
                                   float* __restrict__ scale,
                                   float* __restrict__ shift, int n_nodes) {
  int f = threadIdx.x;
  if (f < N_FEATS) {
    float inv_n = 1.f / (float)n_nodes;
    float m   = sums[f] * inv_n;
    float var = sumsq[f] * inv_n - m * m;      // biased variance (matches ref)
    float sc  = gamma[f] * rsqrtf(var + BN_EPS);
    scale[f] = sc;
    shift[f] = beta[f] - m * sc;
  }
}

// ------------- apply BN into padded h0 [n_nodes x KPAD], pad=0 -------------
__global__ void bn_apply_kernel(const float* __restrict__ x,
                                const float* __restrict__ scale,
                                const float* __restrict__ shift,
                                float* __restrict__ h0, int n_nodes) {
  int idx = blockIdx.x * blockDim.x + threadIdx.x;
  int total = n_nodes * KPAD;
  if (idx >= total) return;
  int n = idx / KPAD;
  int f = idx - n * KPAD;
  h0[idx] = (f < N_FEATS) ? x[(long long)n * N_FEATS + f] * scale[f] + shift[f] : 0.f;
}

// ------------- pad W1 [86x64] -> W1p [88x64] with zero rows ----------------
__global__ void pad_w1_kernel(const float* __restrict__ W1, float* __restrict__ W1p) {
  int idx = blockIdx.x * blockDim.x + threadIdx.x;
  if (idx < KPAD * GCN_H) {
    int k = idx / GCN_H;
    W1p[idx] = (k < N_FEATS) ? W1[idx] : 0.f;
  }
}

// ---------------------------- in-degree (float) ----------------------------
__global__ void degree_kernel(const int* __restrict__ edge_dst,
                              float* __restrict__ deg, int n_edges) {
  int e = blockIdx.x * blockDim.x + threadIdx.x;
  if (e < n_edges) atomicAdd(&deg[edge_dst[e]], 1.0f);
}

// ------------- edge scatter: msg[dst] += h[src] (wave32 per edge) ----------
__global__ void scatter_kernel(const int* __restrict__ src,
                               const int* __restrict__ dst,
                               const float* __restrict__ h,
                               float* __restrict__ msg,
                               int n_edges, int feats, int stride) {
  int wave = (int)((blockIdx.x * blockDim.x + threadIdx.x) >> 5);
  int lane = threadIdx.x & 31;
  if (wave >= n_edges) return;
  int s = src[wave];
  int d = dst[wave];
  const float* hs = h + (long long)s * stride;
  float* md = msg + (long long)d * stride;
  for (int f = lane; f < feats; f += 32)
    atomicAdd(&md[f], hs[f]);
}

// ------- mean finalize (in place): msg[n,k] = deg>0 ? msg/deg : h[n,k] -----
template <int S>
__global__ void agg_finalize_kernel(float* __restrict__ msg,
                                    const float* __restrict__ hin,
                                    const float* __restrict__ deg, int n_nodes) {
  int idx = blockIdx.x * blockDim.x + threadIdx.x;
  if (idx >= n_nodes * S) return;
  int n = idx / S;
  float d = deg[n];
  float v = msg[idx];
  float hv = hin[idx];
  // branchless select: avoids EXEC divergence
  msg[idx] = (d > 0.f) ? v * __frcp_rn(d) : hv;
}

// ----------- GEMM (WMMA f32 16x16x4) + bias + ReLU over A[n x K] -----------
// One wave computes a 16-row x 64-col tile of relu(A @ W + b).
template <int K>
__global__ void gcn_gemm_kernel(const float* __restrict__ A,
                                const float* __restrict__ W,
                                const float* __restrict__ bias,
                                float* __restrict__ out,
                                int n_nodes) {
  int wavesPerBlock = blockDim.x >> 5;
  int wave = blockIdx.x * wavesPerBlock + (int)(threadIdx.x >> 5);
  int lane = threadIdx.x & 31;
  int mtiles = (n_nodes + 15) >> 4;
  if (wave >= mtiles) return;           // wave-uniform exit: EXEC stays full
  int mbase = wave << 4;

  // A fragment addressing (ISA 32-bit A 16x4): lane&15 = M row,
  // lanes>=16 hold K+2/K+3 in VGPR0/1.
  int mrow = mbase + (lane & 15);
  if (mrow > n_nodes - 1) mrow = n_nodes - 1;     // clamp loads (EXEC all-1)
  int khalf = (lane >> 4) << 1;                   // 0 or 2
  const float* aR = A + (long long)mrow * K + khalf;
  int ncol = lane & 15;                           // B/C/D column within tile

  v8f c0 = {}, c1 = {}, c2 = {}, c3 = {};
#pragma unroll
  for (int kk = 0; kk < K; kk += 4) {
    v2f a = *(const v2f*)(aR + kk);               // aligned b64 load
    const float* w0 = W + (kk + khalf) * GCN_H + ncol;
    v2f b0, b1, b2, b3;
    b0.x = w0[0];   b0.y = w0[GCN_H];
    b1.x = w0[16];  b1.y = w0[GCN_H + 16];
    b2.x = w0[32];  b2.y = w0[GCN_H + 32];
    b3.x = w0[48];  b3.y = w0[GCN_H + 48];
    c0 = __builtin_amdgcn_wmma_f32_16x16x4_f32(false, a, false, b0, (short)0, c0, false, false);
    c1 = __builtin_amdgcn_wmma_f32_16x16x4_f32(false, a, false, b1, (short)0, c1, false, false);
    c2 = __builtin_amdgcn_wmma_f32_16x16x4_f32(false, a, false, b2, (short)0, c2, false, false);
    c3 = __builtin_amdgcn_wmma_f32_16x16x4_f32(false, a, false, b3, (short)0, c3, false, false);
  }

  // C/D layout: VGPR r -> M = r (lanes 0-15) or 8+r (lanes 16-31); lane&15 = N.
  int rowOff = (lane >> 4) << 3;                  // 0 or 8
  float bb0 = bias[ncol], bb1 = bias[ncol + 16], bb2 = bias[ncol + 32], bb3 = bias[ncol + 48];
#pragma unroll
  for (int r = 0; r < 8; ++r) {
    int row = mbase + rowOff + r;
    if (row < n_nodes) {
      long long o = (long long)row * GCN_H + ncol;
      out[o]      = fmaxf(c0[r] + bb0, 0.f);
      out[o + 16] = fmaxf(c1[r] + bb1, 0.f);
      out[o + 32] = fmaxf(c2[r] + bb2, 0.f);
      out[o + 48] = fmaxf(c3[r] + bb3, 0.f);
    }
  }
}

// --------- per-graph mean pool (sorted graph_ids) + FC head + sigmoid ------
__global__ void pool_fc_kernel(const float* __restrict__ h2,
                               const int* __restrict__ gids,
                               const float* __restrict__ fc1w,
                               const float* __restrict__ fc1b,
                               const float* __restrict__ fc2w,
                               const float* __restrict__ fc2b,
                               float* __restrict__ out, int n_nodes) {
  __shared__ float red[256];
  __shared__ float hg[GCN_H];
  __shared__ float a1[FC_H];
  int g = blockIdx.x;
  int tid = threadIdx.x;
  // binary search [start, end) of graph g in sorted gids
  int lo = 0, hi = n_nodes;
  while (lo < hi) { int mid = (lo + hi) >> 1; if (gids[mid] < g) lo = mid + 1; else hi = mid; }
  int start = lo;
  hi = n_nodes;
  while (lo < hi) { int mid = (lo + hi) >> 1; if (gids[mid] < g + 1) lo = mid + 1; else hi = mid; }
  int end = lo;
  int cnt = end - start;

  int f = tid & 63, grp = tid >> 6;               // 4 row-groups x 64 feats
  float p = 0.f;
  for (int r = start + grp; r < end; r += 4)
    p += h2[(long long)r * GCN_H + f];
  red[tid] = p;
  __syncthreads();
  if (tid < GCN_H) {
    float s = red[tid] + red[tid + 64] + red[tid + 128] + red[tid + 192];
    hg[tid] = s / fmaxf((float)cnt, 1.f);
  }
  __syncthreads();
  if (tid < FC_H) {
    float s = fc1b[tid];
    for (int k = 0; k < GCN_H; ++k) s += hg[k] * fc1w[k * FC_H + tid];
    a1[tid] = s;
  }
  __syncthreads();
  if (tid == 0) {
    float s = fc2b[0];
    for (int j = 0; j < FC_H; ++j) s += a1[j] * fc2w[j];
    out[g] = 1.f / (1.f + expf(-s));
  }
}

// ---------------------------------------------------------------------------
extern "C" void kernel_launch(void* const* d_in, const int* in_sizes, int n_in,
                              void* d_out, int out_size, void* d_ws, size_t ws_size,
                              hipStream_t stream) {
  const float* x     = (const float*)d_in[0];
  const int*   esrc  = (const int*)d_in[1];
  const int*   edst  = (const int*)d_in[2];
  const int*   gids  = (const int*)d_in[3];
  // d_in[4] = num_graphs scalar (== out_size)
  const float* gamma = (const float*)d_in[5];
  const float* beta  = (const float*)d_in[6];
  const float* W1    = (const float*)d_in[7];
  const float* b1    = (const float*)d_in[8];
  const float* W2    = (const float*)d_in[9];
  const float* b2    = (const float*)d_in[10];
  const float* fc1w  = (const float*)d_in[11];
  const float* fc1b  = (const float*)d_in[12];
  const float* fc2w  = (const float*)d_in[13];
  const float* fc2b  = (const float*)d_in[14];

  int n_nodes  = in_sizes[0] / N_FEATS;   // 100000
  int n_edges  = in_sizes[1];             // 1600000
  int n_graphs = out_size;                // 64

  float* ws = (float*)d_ws;
  size_t off = 0;
  float* h0    = ws + off; off += (size_t)n_nodes * KPAD;   // BN out (stride 88); reused as h2 (stride 64)
  float* msg   = ws + off; off += (size_t)n_nodes * KPAD;   // scatter buffer (reused both layers)
  float* h1    = ws + off; off += (size_t)n_nodes * GCN_H;
  float* deg   = ws + off; off += (size_t)n_nodes;
  float* sums  = ws + off; off += N_FEATS;
  float* sumsq = ws + off; off += N_FEATS;
  float* scale = ws + off; off += N_FEATS;
  float* shift = ws + off; off += N_FEATS;
  float* W1p   = ws + off; off += (size_t)KPAD * GCN_H;
  (void)ws_size; (void)n_in;

  // zero deg + sums + sumsq (contiguous region)
  hipMemsetAsync(deg, 0, (size_t)(n_nodes + 2 * N_FEATS) * sizeof(float), stream);

  bn_stats_kernel<<<512, 256, 0, stream>>>(x, sums, sumsq, n_nodes);
  bn_finalize_kernel<<<1, 128, 0, stream>>>(sums, sumsq, gamma, beta, scale, shift, n_nodes);
  {
    int total = n_nodes * KPAD;
    bn_apply_kernel<<<(total + 255) / 256, 256, 0, stream>>>(x, scale, shift, h0, n_nodes);
  }
  pad_w1_kernel<<<(KPAD * GCN_H + 255) / 256, 256, 0, stream>>>(W1, W1p);
  degree_kernel<<<(n_edges + 255) / 256, 256, 0, stream>>>(edst, deg, n_edges);

  int mtiles = (n_nodes + 15) / 16;
  int gemm_blocks = (mtiles + 7) / 8;               // 8 waves per 256-thread block
  int scat_blocks = (int)(((long long)n_edges * 32 + 255) / 256);
  int fin1_blocks = (n_nodes * KPAD + 255) / 256;
  int fin2_blocks = (n_nodes * GCN_H + 255) / 256;

  // ----- GCN layer 1: agg over 86 BN features (stride KPAD), W1 [88x64] ----
  hipMemsetAsync(msg, 0, (size_t)n_nodes * KPAD * sizeof(float), stream);
  scatter_kernel<<<scat_blocks, 256, 0, stream>>>(esrc, edst, h0, msg, n_edges, N_FEATS, KPAD);
  agg_finalize_kernel<KPAD><<<fin1_blocks, 256, 0, stream>>>(msg, h0, deg, n_nodes);
  gcn_gemm_kernel<KPAD><<<gemm_blocks, 256, 0, stream>>>(msg, W1p, b1, h1, n_nodes);

  // ----- GCN layer 2: agg over 64 features (stride 64), W2 [64x64] ---------
  float* h2 = h0;                                   // reuse h0 region
  hipMemsetAsync(msg, 0, (size_t)n_nodes * GCN_H * sizeof(float), stream);
  scatter_kernel<<<scat_blocks, 256, 0, stream>>>(esrc, edst, h1, msg, n_edges, GCN_H, GCN_H);
  agg_finalize_kernel<GCN_H><<<fin2_blocks, 256, 0, stream>>>(msg, h1, deg, n_nodes);
  gcn_gemm_kernel<GCN_H><<<gemm_blocks, 256, 0, stream>>>(msg, W2, b2, h2, n_nodes);

  // ----- pool + FC head ----------------------------------------------------
  pool_fc_kernel<<<n_graphs, 256, 0, stream>>>(h2, gids, fc1w, fc1b, fc2w, fc2b,
                                               (float*)d_out, n_nodes);
}